// L2BoundedLTICell_34857954574987
// MI455X (gfx1250) — compile-verified
//
#include <hip/hip_runtime.h>
#include <hip/hip_bf16.h>
#include <math.h>

// ---------------------------------------------------------------------------
// L2BoundedLTICell on MI455X (gfx1250): fp32 WMMA (v_wmma_f32_16x16x4_f32)
// + gfx1250 async global->LDS staging (ASYNCcnt) in the hot kernels.
// Chunked parallel scan (64 chunks x 64 steps) + WMMA GEMMs for setup/output.
// Workspace requirement: ~18 MB.
// ---------------------------------------------------------------------------

typedef float v2f __attribute__((ext_vector_type(2)));
typedef float v8f __attribute__((ext_vector_type(8)));

#define N_    512      // state / H / P
#define Bsz   16       // batch
#define T_    4096     // time steps
#define CH    64       // chunks
#define CL    64       // chunk length
#define PAD   18       // LDS row stride (even -> b64-aligned frags, 64-bank clean)
#define NK    1024     // K_raw dim

__device__ __forceinline__ v8f wmma_f32_k4(v2f a, v2f b, v8f c) {
  // D = A(16x4) * B(4x16) + C(16x16), fp32
  return __builtin_amdgcn_wmma_f32_16x16x4_f32(
      /*neg_a=*/false, a, /*neg_b=*/false, b,
      /*c_mod=*/(short)0, c, /*reuse_a=*/false, /*reuse_b=*/false);
}

// A-fragment (16x4 fp32) from LDS, row-major [row][k], row stride st (even).
// lanes 0-15: rows m0..m0+15 at K = k0,k0+1 ; lanes 16-31: same rows at K+2.
// 8B-aligned -> single ds_load_b64.
__device__ __forceinline__ v2f lds_afrag(const float* s, int m0, int k0, int st) {
  int l = threadIdx.x & 31;
  int row = m0 + (l & 15);
  int kk  = k0 + 2 * (l >> 4);
  return *(const v2f*)(s + row * st + kk);
}

// B-fragment (4x16 fp32) from LDS, [k][col] with row stride st.
__device__ __forceinline__ v2f lds_bfrag(const float* s, int k0, int n0, int st) {
  int l = threadIdx.x & 31;
  int col = n0 + (l & 15);
  int kk  = k0 + 2 * (l >> 4);
  v2f b;
  b.x = s[kk * st + col];
  b.y = s[(kk + 1) * st + col];
  return b;
}

__device__ __forceinline__ void v8f_zero(v8f& v) {
#pragma unroll
  for (int r = 0; r < 8; ++r) v[r] = 0.f;
}

// --------- gfx1250 async global->LDS copies (tracked by ASYNCcnt) ----------
__device__ __forceinline__ unsigned lds_off32(const void* p) {
  // low 32 bits of a generic pointer into LDS == LDS byte offset
  return (unsigned)(unsigned long long)p;
}
__device__ __forceinline__ void async_g2lds_b64(void* lds_dst, const float* g) {
  asm volatile("global_load_async_to_lds_b64 %0, %1, off"
               :: "v"(lds_off32(lds_dst)), "v"(g) : "memory");
}
__device__ __forceinline__ void async_g2lds_b32(void* lds_dst, const float* g) {
  asm volatile("global_load_async_to_lds_b32 %0, %1, off"
               :: "v"(lds_off32(lds_dst)), "v"(g) : "memory");
}
__device__ __forceinline__ void wait_async0() {
  asm volatile("s_wait_asynccnt 0" ::: "memory");
}

// Stage a 512x16 k-slab M[:, kb*16 .. +16] into sSlab (stride PAD) via async b64.
__device__ __forceinline__ void stage_slab_async(float* sSlab, const float* M,
                                                 int kb, int tid) {
  // 4096 dword-pairs, 16 per thread
  for (int p = tid; p < 4096; p += 256) {
    int row = p >> 3;
    int kk2 = (p & 7) * 2;
    async_g2lds_b64(&sSlab[row * PAD + kk2], M + (size_t)row * N_ + kb * 16 + kk2);
  }
}

// ---------------------------------------------------------------------------
// Generic WMMA GEMM: D = alpha*(op(A) @ B) + beta*C   (row-major, mult of 64)
// alpha = alpha_mul * (*alpha_ptr)  (alpha_ptr may be null -> 1.0)
// ---------------------------------------------------------------------------
__global__ void __launch_bounds__(256)
gemm_kernel(int M, int Nn, int K,
            const float* __restrict__ A, int lda, int transA,
            const float* __restrict__ B, int ldb,
            const float* __restrict__ Cc, int ldc, float beta,
            float* __restrict__ D, int ldd,
            const float* __restrict__ alpha_ptr, float alpha_mul) {
  __shared__ float sA[64 * PAD];   // 64 rows x 16 k
  __shared__ float sB[16 * 68];    // 16 k x 64 cols (stride 68)
  const int tid = threadIdx.x;
  const int w   = tid >> 5;
  const int l   = tid & 31;
  const int m0  = blockIdx.y * 64;
  const int n0  = blockIdx.x * 64;
  const int sm  = w >> 1;          // 0..3  row sub-tile
  const int sn0 = (w & 1) * 2;     // 0 or 2, wave owns sn0,sn0+1
  const float alpha = alpha_mul * (alpha_ptr ? *alpha_ptr : 1.f);

  v8f acc[2];
  v8f_zero(acc[0]);
  v8f_zero(acc[1]);

  const int nkb = K >> 4;
  for (int kb = 0; kb < nkb; ++kb) {
    __syncthreads();
    for (int i = tid; i < 1024; i += 256) {     // stage A tile 64x16
      int r = i >> 4, kk = i & 15;
      float v = transA ? A[(size_t)(kb * 16 + kk) * lda + (m0 + r)]
                       : A[(size_t)(m0 + r) * lda + kb * 16 + kk];
      sA[r * PAD + kk] = v;
    }
    for (int i = tid; i < 1024; i += 256) {     // stage B tile 16x64
      int kk = i >> 6, c = i & 63;
      sB[kk * 68 + c] = B[(size_t)(kb * 16 + kk) * ldb + n0 + c];
    }
    __syncthreads();
#pragma unroll
    for (int k4 = 0; k4 < 16; k4 += 4) {
      v2f af = lds_afrag(sA, sm * 16, k4, PAD);
#pragma unroll
      for (int s = 0; s < 2; ++s) {
        v2f bf = lds_bfrag(sB, k4, (sn0 + s) * 16, 68);
        acc[s] = wmma_f32_k4(af, bf, acc[s]);
      }
    }
  }
  // epilogue
#pragma unroll
  for (int s = 0; s < 2; ++s) {
    int cn = n0 + (sn0 + s) * 16 + (l & 15);
    int rb = m0 + sm * 16 + 8 * (l >> 4);
#pragma unroll
    for (int r = 0; r < 8; ++r) {
      float v = alpha * acc[s][r];
      if (beta != 0.f) v += beta * Cc[(size_t)(rb + r) * ldc + cn];
      D[(size_t)(rb + r) * ldd + cn] = v;
    }
  }
}

// ---------------------------------------------------------------------------
// Power iteration on M = K^T K (1024x1024, symmetric) -> sigma = ||K||_2.
// Single workgroup, M stays in L2. Also computes gamma = exp(log_gamma).
// scal[0]=sigma scal[1]=1/(sigma+0.002) scal[2]=gamma scal[3]=gamma*scal[1]
// ---------------------------------------------------------------------------
__global__ void __launch_bounds__(512)
power_iter_kernel(const float* __restrict__ Mk, const float* __restrict__ log_gamma,
                  float* __restrict__ scal) {
  __shared__ float sv[1024];
  __shared__ float red[512];
  const int tid = threadIdx.x;
  for (int i = tid; i < 1024; i += 512) {
    unsigned h = (unsigned)i * 2654435761u + 12345u;
    h ^= h >> 16; h *= 2246822519u; h ^= h >> 13;
    sv[i] = ((h & 0xFFFF) * (1.f / 65535.f)) - 0.45f;   // deterministic pseudo-random
  }
  __syncthreads();
  float lambda = 0.f;
  for (int it = 0; it < 64; ++it) {
    float w0 = 0.f, w1 = 0.f;
    const float* r0 = Mk + (size_t)tid * 1024;
    const float* r1 = Mk + (size_t)(tid + 512) * 1024;
    for (int k = 0; k < 1024; ++k) {
      float vk = sv[k];
      w0 = fmaf(r0[k], vk, w0);
      w1 = fmaf(r1[k], vk, w1);
    }
    red[tid] = w0 * w0 + w1 * w1;
    __syncthreads();
    for (int s = 256; s > 0; s >>= 1) {
      if (tid < s) red[tid] += red[tid + s];
      __syncthreads();
    }
    float nrm = sqrtf(red[0]);
    lambda = nrm;                       // ||M v|| with unit v -> lambda_max(M)
    float inv = (nrm > 0.f) ? 1.f / nrm : 0.f;
    __syncthreads();
    sv[tid] = w0 * inv;
    sv[tid + 512] = w1 * inv;
    __syncthreads();
  }
  if (tid == 0) {
    float sigma = fmaxf(sqrtf(lambda), 1e-5f);
    float inv_s = 1.f / (sigma + 0.002f);
    float gamma = expf(log_gamma[0]);
    scal[0] = sigma;
    scal[1] = inv_s;
    scal[2] = gamma;
    scal[3] = gamma * inv_s;
  }
}

__global__ void set_identity(float* __restrict__ X, int n) {
  int i = blockIdx.x * 256 + threadIdx.x;
  if (i < n * n) X[i] = ((i / n) == (i % n)) ? 1.f : 0.f;
}

__global__ void dscale_kernel(const float* __restrict__ K_raw,
                              const float* __restrict__ scal,
                              float* __restrict__ Dm) {
  int i = blockIdx.x * 256 + threadIdx.x;
  if (i < N_ * N_) {
    int r = i >> 9, c = i & 511;
    Dm[i] = scal[3] * K_raw[(size_t)(N_ + r) * NK + N_ + c];
  }
}

// states[b][0][:] = state_in ; bnd[0][n][b] = state_in[b][n]
__global__ void init_states_kernel(const float* __restrict__ state_in,
                                   float* __restrict__ states,
                                   float* __restrict__ bnd0) {
  int i = blockIdx.x * 256 + threadIdx.x;
  if (i < Bsz * N_) {
    int b = i >> 9, n = i & 511;
    float v = state_in[i];
    states[(size_t)b * ((T_ + 1) * (size_t)N_) + n] = v;
    bnd0[n * Bsz + b] = v;
  }
}

// ---------------------------------------------------------------------------
// Chunked scan.  One workgroup (256 thr / 8 waves) per chunk of 64 steps.
// Per step: Ynew(512x16) = A@Ycur + B@u_t, all via wmma_f32_16x16x4.
// Slab + u_t staging use gfx1250 async global->LDS copies.
// phase1: bnd==null (zero init), writes chunk_end[chunk].
// phase3: bnd!=null (boundary init), writes states[b][t+1][n].
// ---------------------------------------------------------------------------
__global__ void __launch_bounds__(256)
chunk_scan_kernel(const float* __restrict__ Amat,
                  const float* __restrict__ Bmat,
                  const float* __restrict__ u,
                  const float* __restrict__ bnd,
                  float* __restrict__ chunk_end,
                  float* __restrict__ states_out) {
  __shared__ float sY0[N_ * PAD];
  __shared__ float sY1[N_ * PAD];
  __shared__ float sSlab[N_ * PAD];
  __shared__ float sU[N_ * PAD];
  float* yc = sY0;
  float* yn = sY1;
  const int tid = threadIdx.x;
  const int w   = tid >> 5;
  const int l   = tid & 31;
  const int chunk = blockIdx.x;

  if (bnd) {
    const float* b0 = bnd + (size_t)chunk * (N_ * Bsz);
    for (int i = tid; i < N_ * Bsz; i += 256) {
      int n = i >> 4, b = i & 15;
      yc[n * PAD + b] = b0[i];
    }
  } else {
    for (int i = tid; i < N_ * Bsz; i += 256) {
      int n = i >> 4, b = i & 15;
      yc[n * PAD + b] = 0.f;
    }
  }
  __syncthreads();

  for (int step = 0; step < CL; ++step) {
    const int t = chunk * CL + step;
    // async transpose-stage u_t : sU[h][b] = u[b][t][h]
    for (int i = tid; i < N_ * Bsz; i += 256) {
      int b = i >> 9, h = i & 511;
      async_g2lds_b32(&sU[h * PAD + b],
                      u + (size_t)b * (T_ * (size_t)N_) + (size_t)t * N_ + h);
    }
    v8f acc[4];
#pragma unroll
    for (int j = 0; j < 4; ++j) v8f_zero(acc[j]);

    // pass 0: acc += Bmat @ u_t ; pass 1: acc += Amat @ Ycur
    for (int p = 0; p < 2; ++p) {
      const float* M   = p ? Amat : Bmat;
      const float* rhs = p ? yc : sU;
      for (int kb = 0; kb < 32; ++kb) {
        __syncthreads();                     // prior consumers done
        stage_slab_async(sSlab, M, kb, tid); // async 512x16 k-slab of M
        wait_async0();
        __syncthreads();                     // all waves' copies visible
#pragma unroll
        for (int k4 = 0; k4 < 16; k4 += 4) {
          v2f bf = lds_bfrag(rhs, kb * 16 + k4, 0, PAD);
#pragma unroll
          for (int j = 0; j < 4; ++j) {
            int m0 = (w * 4 + j) * 16;
            v2f af = lds_afrag(sSlab, m0, k4, PAD);
            acc[j] = wmma_f32_k4(af, bf, acc[j]);
          }
        }
      }
    }
    __syncthreads();
    // write Ynext to LDS (C/D layout: VGPR r -> row r / r+8, lane -> col)
#pragma unroll
    for (int j = 0; j < 4; ++j) {
      int m0 = (w * 4 + j) * 16;
      int col = l & 15;
      int rbase = m0 + 8 * (l >> 4);
#pragma unroll
      for (int r = 0; r < 8; ++r) yn[(rbase + r) * PAD + col] = acc[j][r];
    }
    __syncthreads();
    if (states_out) {   // coalesced store of states[b][t+1][:]
      for (int i = tid; i < N_ * Bsz; i += 256) {
        int b = i >> 9, n = i & 511;
        states_out[(size_t)b * ((T_ + 1) * (size_t)N_) + (size_t)(t + 1) * N_ + n] =
            yn[n * PAD + b];
      }
    }
    float* tmp = yc; yc = yn; yn = tmp;
    __syncthreads();
  }

  if (chunk_end) {
    float* ce = chunk_end + (size_t)chunk * (N_ * Bsz);
    for (int i = tid; i < N_ * Bsz; i += 256) {
      int n = i >> 4, b = i & 15;
      ce[i] = yc[n * PAD + b];
    }
  }
}

// ---------------------------------------------------------------------------
// Phase 2: sequential boundary scan over 64 chunks (single workgroup).
// bnd[j+1] = G @ bnd[j] + chunk_end[j],  G = A^64 (L2-resident).
// ---------------------------------------------------------------------------
__global__ void __launch_bounds__(256)
boundary_kernel(const float* __restrict__ G,
                const float* __restrict__ chunk_end,
                float* __restrict__ bnd) {
  __shared__ float sY[N_ * PAD];
  __shared__ float sSlab[N_ * PAD];
  const int tid = threadIdx.x;
  const int w   = tid >> 5;
  const int l   = tid & 31;

  for (int i = tid; i < N_ * Bsz; i += 256) {   // load bnd[0]
    int n = i >> 4, b = i & 15;
    sY[n * PAD + b] = bnd[i];
  }
  __syncthreads();

  for (int j = 0; j < CH - 1; ++j) {
    v8f acc[4];
    const float* ce = chunk_end + (size_t)j * (N_ * Bsz);
    {
      int col = l & 15;
#pragma unroll
      for (int jj = 0; jj < 4; ++jj) {
        int rbase = (w * 4 + jj) * 16 + 8 * (l >> 4);
#pragma unroll
        for (int r = 0; r < 8; ++r) acc[jj][r] = ce[(rbase + r) * Bsz + col];
      }
    }
    for (int kb = 0; kb < 32; ++kb) {
      __syncthreads();
      stage_slab_async(sSlab, G, kb, tid);
      wait_async0();
      __syncthreads();
#pragma unroll
      for (int k4 = 0; k4 < 16; k4 += 4) {
        v2f bf = lds_bfrag(sY, kb * 16 + k4, 0, PAD);
#pragma unroll
        for (int jj = 0; jj < 4; ++jj) {
          int m0 = (w * 4 + jj) * 16;
          v2f af = lds_afrag(sSlab, m0, k4, PAD);
          acc[jj] = wmma_f32_k4(af, bf, acc[jj]);
        }
      }
    }
    __syncthreads();
#pragma unroll
    for (int jj = 0; jj < 4; ++jj) {   // overwrite sY with new boundary
      int col = l & 15;
      int rbase = (w * 4 + jj) * 16 + 8 * (l >> 4);
#pragma unroll
      for (int r = 0; r < 8; ++r) sY[(rbase + r) * PAD + col] = acc[jj][r];
    }
    __syncthreads();
    float* bo = bnd + (size_t)(j + 1) * (N_ * Bsz);
    for (int i = tid; i < N_ * Bsz; i += 256) {
      int n = i >> 4, b = i & 15;
      bo[i] = sY[n * PAD + b];
    }
    __syncthreads();
  }
}

// ---------------------------------------------------------------------------
// Output: out[b][t][:] = Cm @ states[b][t][:] + Dm @ u[b][t][:]
// One workgroup per t; async staging; LDS-transposed coalesced store.
// ---------------------------------------------------------------------------
__global__ void __launch_bounds__(256)
output_kernel(const float* __restrict__ Cm, const float* __restrict__ Dm,
              const float* __restrict__ states, const float* __restrict__ u,
              float* __restrict__ out) {
  __shared__ float sX[N_ * PAD];
  __shared__ float sU[N_ * PAD];
  __shared__ float sSlab[N_ * PAD];
  const int tid = threadIdx.x;
  const int w   = tid >> 5;
  const int l   = tid & 31;
  const int t   = blockIdx.x;

  for (int i = tid; i < N_ * Bsz; i += 256) {   // async transpose-stage u_t, x_t
    int b = i >> 9, n = i & 511;
    async_g2lds_b32(&sU[n * PAD + b],
                    u + (size_t)b * (T_ * (size_t)N_) + (size_t)t * N_ + n);
    async_g2lds_b32(&sX[n * PAD + b],
                    states + (size_t)b * ((T_ + 1) * (size_t)N_) + (size_t)t * N_ + n);
  }
  v8f acc[4];
#pragma unroll
  for (int j = 0; j < 4; ++j) v8f_zero(acc[j]);

  for (int p = 0; p < 2; ++p) {
    const float* M   = p ? Cm : Dm;
    const float* rhs = p ? sX : sU;
    for (int kb = 0; kb < 32; ++kb) {
      __syncthreads();
      stage_slab_async(sSlab, M, kb, tid);
      wait_async0();
      __syncthreads();
#pragma unroll
      for (int k4 = 0; k4 < 16; k4 += 4) {
        v2f bf = lds_bfrag(rhs, kb * 16 + k4, 0, PAD);
#pragma unroll
        for (int j = 0; j < 4; ++j) {
          int m0 = (w * 4 + j) * 16;
          v2f af = lds_afrag(sSlab, m0, k4, PAD);
          acc[j] = wmma_f32_k4(af, bf, acc[j]);
        }
      }
    }
  }
  __syncthreads();
#pragma unroll
  for (int j = 0; j < 4; ++j) {       // reuse sSlab as [n][b] transpose buffer
    int col = l & 15;
    int rbase = (w * 4 + j) * 16 + 8 * (l >> 4);
#pragma unroll
    for (int r = 0; r < 8; ++r) sSlab[(rbase + r) * PAD + col] = acc[j][r];
  }
  __syncthreads();
  for (int i = tid; i < N_ * Bsz; i += 256) {
    int b = i >> 9, n = i & 511;
    out[(size_t)b * (T_ * (size_t)N_) + (size_t)t * N_ + n] = sSlab[n * PAD + b];
  }
}

// ---------------------------------------------------------------------------
// Host-side orchestration (graph-capture safe: stream-only, no sync/alloc).
// ---------------------------------------------------------------------------
extern "C" void kernel_launch(void* const* d_in, const int* in_sizes, int n_in,
                              void* d_out, int out_size, void* d_ws, size_t ws_size,
                              hipStream_t stream) {
  (void)in_sizes; (void)n_in; (void)out_size; (void)ws_size;
  const float* u         = (const float*)d_in[0];
  const float* state_in  = (const float*)d_in[1];
  const float* S         = (const float*)d_in[2];
  const float* K_raw     = (const float*)d_in[3];
  const float* log_gamma = (const float*)d_in[4];

  float* out_ptr    = (float*)d_out;                                // 16*4096*512
  float* states_ptr = (float*)d_out + (size_t)Bsz * T_ * N_;        // 16*4097*512

  // workspace layout (floats); total ~4.47M floats (~18 MB)
  float* ws = (float*)d_ws;
  const size_t MAT = (size_t)N_ * N_;           // 262144
  size_t off = 0;
  float* scal = ws + off; off += 16;
  float* KTK  = ws + off; off += (size_t)NK * NK;   // 1048576
  float* X0   = ws + off; off += MAT;           // Newton ping
  float* X1   = ws + off; off += MAT;           // Newton pong
  float* MT   = ws + off; off += MAT;           // temp
  float* AM   = ws + off; off += MAT;           // A
  float* BM   = ws + off; off += MAT;           // B
  float* CM   = ws + off; off += MAT;           // C
  float* DM   = ws + off; off += MAT;           // D
  float* GA   = ws + off; off += MAT;           // A^(2k) ping
  float* GB   = ws + off; off += MAT;           // A^(2k) pong
  float* BND  = ws + off; off += (size_t)(CH + 1) * N_ * Bsz;   // boundaries
  float* CE   = ws + off; off += (size_t)CH * N_ * Bsz;         // chunk-end

  const dim3 blk(256);
  const dim3 g512(8, 8);     // 512x512 output tiles of 64
  const dim3 g1024(16, 16);

  // 1) KtK = K^T K  (for spectral norm)
  gemm_kernel<<<g1024, blk, 0, stream>>>(NK, NK, NK, K_raw, NK, 1, K_raw, NK,
                                         nullptr, 0, 0.f, KTK, NK, nullptr, 1.f);
  // 2) sigma, gamma, scale factors
  power_iter_kernel<<<1, 512, 0, stream>>>(KTK, log_gamma, scal);

  // 3) Sinv via Newton-Schulz: X <- 2X - X(SX), X0 = I, 8 iters
  set_identity<<<(int)((MAT + 255) / 256), blk, 0, stream>>>(X0, N_);
  float* Xc = X0; float* Xn = X1;
  for (int it = 0; it < 8; ++it) {
    gemm_kernel<<<g512, blk, 0, stream>>>(N_, N_, N_, S, N_, 0, Xc, N_,
                                          nullptr, 0, 0.f, MT, N_, nullptr, 1.f);
    gemm_kernel<<<g512, blk, 0, stream>>>(N_, N_, N_, Xc, N_, 0, MT, N_,
                                          Xc, N_, 2.f, Xn, N_, nullptr, -1.f);
    float* t = Xc; Xc = Xn; Xn = t;
  }
  // Xc == X0 holds Sinv after 8 iterations.

  // 4) A = Sinv @ (K11/s) @ S ; B = gamma/s * Sinv @ K12 ; C = 1/s * K21 @ S
  gemm_kernel<<<g512, blk, 0, stream>>>(N_, N_, N_, Xc, N_, 0, K_raw, NK,
                                        nullptr, 0, 0.f, MT, N_, scal + 1, 1.f);
  gemm_kernel<<<g512, blk, 0, stream>>>(N_, N_, N_, MT, N_, 0, S, N_,
                                        nullptr, 0, 0.f, AM, N_, nullptr, 1.f);
  gemm_kernel<<<g512, blk, 0, stream>>>(N_, N_, N_, Xc, N_, 0, K_raw + N_, NK,
                                        nullptr, 0, 0.f, BM, N_, scal + 3, 1.f);
  gemm_kernel<<<g512, blk, 0, stream>>>(N_, N_, N_, K_raw + (size_t)N_ * NK, NK, 0,
                                        S, N_, nullptr, 0, 0.f, CM, N_, scal + 1, 1.f);
  dscale_kernel<<<(int)((MAT + 255) / 256), blk, 0, stream>>>(K_raw, scal, DM);

  // 5) G = A^64 via 6 squarings (AM->GA->GB->GA->GB->GA->GB)
  gemm_kernel<<<g512, blk, 0, stream>>>(N_, N_, N_, AM, N_, 0, AM, N_,
                                        nullptr, 0, 0.f, GA, N_, nullptr, 1.f);
  float* Gc = GA; float* Gn = GB;
  for (int i = 0; i < 5; ++i) {
    gemm_kernel<<<g512, blk, 0, stream>>>(N_, N_, N_, Gc, N_, 0, Gc, N_,
                                          nullptr, 0, 0.f, Gn, N_, nullptr, 1.f);
    float* t = Gc; Gc = Gn; Gn = t;
  }
  // Gc == GB holds A^64.

  // 6) init states[:,0,:] and bnd[0]
  init_states_kernel<<<32, blk, 0, stream>>>(state_in, states_ptr, BND);

  // 7) phase 1: local scans (zero init) -> chunk_end
  chunk_scan_kernel<<<CH, blk, 0, stream>>>(AM, BM, u, nullptr, CE, nullptr);
  // 8) phase 2: boundary scan  bnd[j+1] = G bnd[j] + chunk_end[j]
  boundary_kernel<<<1, blk, 0, stream>>>(Gc, CE, BND);
  // 9) phase 3: local scans with true initial states -> states[:,1..T,:]
  chunk_scan_kernel<<<CH, blk, 0, stream>>>(AM, BM, u, BND, nullptr, states_ptr);
  // 10) output = C @ prestates + D @ u
  output_kernel<<<T_, blk, 0, stream>>>(CM, DM, states_ptr, u, out_ptr);
}